// CellComplexOnline_54065048322397
// MI455X (gfx1250) — compile-verified
//
#include <hip/hip_runtime.h>
#include <math.h>

#define HDIM 128
#define CDIM 64

typedef float v2f __attribute__((ext_vector_type(2)));
typedef float v8f __attribute__((ext_vector_type(8)));

// ---------------------------------------------------------------------------
// fill: set n floats to v
// ---------------------------------------------------------------------------
__global__ __launch_bounds__(256) void fill_kernel(float* __restrict__ p, float v, long n) {
  long i = (long)blockIdx.x * blockDim.x + threadIdx.x;
  if (i < n) p[i] = v;
}

// ---------------------------------------------------------------------------
// WMMA fp32 GEMM: C[M x 128] = A[M x K] @ B[K x 128] + bias (+ optional PReLU)
// One wave computes one 16x16 tile using V_WMMA_F32_16X16X4_F32.
// Block = 256 threads = 8 waves = one 16-row stripe across all 128 columns.
// Grid.x = M/16 (M divisible by 16). EXEC is all-ones everywhere (WMMA req).
// ---------------------------------------------------------------------------
template <int K, bool PRELU>
__global__ __launch_bounds__(256) void gemm_wmma_f32(
    const float* __restrict__ A, const float* __restrict__ B,
    const float* __restrict__ bias, const float* __restrict__ alpha,
    float* __restrict__ C) {
  const int lane = threadIdx.x & 31;
  const int wave = threadIdx.x >> 5;   // 0..7 -> column tile
  const int hf   = lane >> 4;          // 0: K pair {0,1}, 1: K pair {2,3}
  const int l    = lane & 15;
  const long row0 = (long)blockIdx.x * 16;
  const int  col0 = wave * 16;

  // A-fragment base: lane l holds A[row0+l][k0 + 2*hf + {0,1}] (aligned float2)
  const float* arow = A + (row0 + l) * K + 2 * hf;
  // B-fragment base: lane l holds B[k0 + 2*hf + {0,1}][col0 + l]
  const float* bcol = B + (long)(2 * hf) * HDIM + col0 + l;

  v8f acc = {};
#pragma unroll
  for (int k0 = 0; k0 < K; k0 += 4) {
    v2f a = *(const v2f*)(arow + k0);
    v2f b;
    b.x = bcol[(long)k0 * HDIM];
    b.y = bcol[(long)k0 * HDIM + HDIM];
    acc = __builtin_amdgcn_wmma_f32_16x16x4_f32(
        /*neg_a=*/false, a, /*neg_b=*/false, b,
        /*c_mod=*/(short)0, acc, /*reuse_a=*/false, /*reuse_b=*/false);
  }

  const float bv = bias[col0 + l];
  const float al = PRELU ? alpha[0] : 0.0f;
#pragma unroll
  for (int j = 0; j < 8; ++j) {
    // C/D layout: VGPR j -> row = row0 + j + 8*hf, col = col0 + lane%16
    const long row = row0 + j + 8 * hf;
    float v = acc[j] + bv;
    if (PRELU) v = (v >= 0.0f) ? v : al * v;
    C[row * HDIM + col0 + l] = v;
  }
}

// ---------------------------------------------------------------------------
// COO scatter SpMM: dst[oidx[e]] += scale * val[e] * src[iidx[e]]  (rows of H)
// One wave per nonzero; each lane owns 4 channels (float4 gather, 4 atomics).
// Transpose = swap oidx/iidx at call site. Does NOT zero dst (caller does, or
// accumulates into live dst for the 0.5*B^T contribution).
// ---------------------------------------------------------------------------
__global__ __launch_bounds__(256) void spmm_scatter(
    float* __restrict__ dst, const float* __restrict__ src,
    const int* __restrict__ oidx, const int* __restrict__ iidx,
    const float* __restrict__ val, int nnz, float scale) {
  const int lane  = threadIdx.x & 31;
  const int wpb   = blockDim.x >> 5;
  int e = blockIdx.x * wpb + (threadIdx.x >> 5);
  const int step  = gridDim.x * wpb;
  for (; e < nnz; e += step) {
    const int   o = oidx[e];
    const int   in = iidx[e];
    const float v = val[e] * scale;
    const float4 x = ((const float4*)(src + (long)in * HDIM))[lane];
    float* d = dst + (long)o * HDIM + lane * 4;
    atomicAdd(d + 0, v * x.x);
    atomicAdd(d + 1, v * x.y);
    atomicAdd(d + 2, v * x.z);
    atomicAdd(d + 3, v * x.w);
  }
}

// ---------------------------------------------------------------------------
// gate = sigmoid(comb @ a); out[row] = beta*out[row] + wsum * gate * comb[row]
// One wave per row; float4 per lane; wave32 shuffle reduction for the dot.
// ---------------------------------------------------------------------------
__global__ __launch_bounds__(256) void gate_accum(
    const float* __restrict__ comb, const float* __restrict__ avec,
    float* __restrict__ out, int n, float wsum, int beta) {
  const int lane = threadIdx.x & 31;
  const long row = (long)blockIdx.x * 8 + (threadIdx.x >> 5);
  if (row >= n) return;
  const float4 cv = ((const float4*)(comb + row * HDIM))[lane];
  const float4 av = ((const float4*)avec)[lane];
  float s = cv.x * av.x + cv.y * av.y + cv.z * av.z + cv.w * av.w;
#pragma unroll
  for (int off = 16; off > 0; off >>= 1) s += __shfl_xor(s, off, 32);
  const float g = 1.0f / (1.0f + __expf(-s));
  float* op = out + row * HDIM + lane * 4;
  float4 o;
  if (beta) {
    o.x = op[0] + wsum * g * cv.x;
    o.y = op[1] + wsum * g * cv.y;
    o.z = op[2] + wsum * g * cv.z;
    o.w = op[3] + wsum * g * cv.w;
  } else {
    o.x = wsum * g * cv.x;
    o.y = wsum * g * cv.y;
    o.z = wsum * g * cv.z;
    o.w = wsum * g * cv.w;
  }
  *(float4*)op = o;
}

// ---------------------------------------------------------------------------
// Host orchestration
// ---------------------------------------------------------------------------
struct CooMat {
  const int* r;
  const int* c;
  const float* v;
  int nnz;
};

extern "C" void kernel_launch(void* const* d_in, const int* in_sizes, int n_in,
                              void* d_out, int out_size, void* d_ws, size_t ws_size,
                              hipStream_t stream) {
  (void)n_in; (void)out_size; (void)ws_size;

  const int  N  = in_sizes[0] / CDIM;     // 100000
  const long NH = (long)N * HDIM;

  const float* xs[3] = {(const float*)d_in[0], (const float*)d_in[1], (const float*)d_in[2]};

  CooMat adj[3];
  for (int d = 0; d < 3; ++d) {
    adj[d].r = (const int*)d_in[3 + 3 * d];
    adj[d].c = (const int*)d_in[4 + 3 * d];
    adj[d].v = (const float*)d_in[5 + 3 * d];
    adj[d].nnz = in_sizes[5 + 3 * d];
  }
  CooMat B1 = {(const int*)d_in[12], (const int*)d_in[13], (const float*)d_in[14], in_sizes[14]};
  CooMat B2 = {(const int*)d_in[15], (const int*)d_in[16], (const float*)d_in[17], in_sizes[17]};
  const CooMat* bdown[3] = {nullptr, &B1, &B2};
  const CooMat* bup[3]   = {&B1, &B2, nullptr};

  const float* pW1   = (const float*)d_in[36];
  const float* pb1   = (const float*)d_in[37];
  const float* alpha = (const float*)d_in[38];
  const float* pW2   = (const float*)d_in[39];
  const float* pb2   = (const float*)d_in[40];

  float* out       = (float*)d_out;
  float* h_online  = out;            // [N,H]
  float* h_pred    = out + NH;       // [N,H]
  float* h_target  = out + 2 * NH;   // [N,H]

  float* xt   = (float*)d_ws;        // [N,H] (also reused as h1 in predictor)
  float* comb = xt + NH;             // [N,H]
  float* t1   = comb + NH;           // [N,H]

  const dim3 blk(256);
  const int  gemmGrid  = N / 16;                 // M tiles (N divisible by 16)
  const int  fillGrid  = (int)((NH + 255) / 256);
  const int  gateGrid  = (N + 7) / 8;

  for (int enc = 0; enc < 2; ++enc) {
    const int base = (enc == 0) ? 18 : 27;
    float* enc_out = (enc == 0) ? h_online : h_target;

    for (int d = 0; d < 3; ++d) {
      const float* W  = (const float*)d_in[base + 3 * d + 0];
      const float* bv = (const float*)d_in[base + 3 * d + 1];
      const float* av = (const float*)d_in[base + 3 * d + 2];

      // xt = x_d @ W + b   (fp32 WMMA GEMM, K=64)
      gemm_wmma_f32<CDIM, false><<<gemmGrid, blk, 0, stream>>>(xs[d], W, bv, nullptr, xt);

      // comb = A_d @ xt
      fill_kernel<<<fillGrid, blk, 0, stream>>>(comb, 0.0f, NH);
      spmm_scatter<<<(adj[d].nnz + 7) / 8, blk, 0, stream>>>(
          comb, xt, adj[d].r, adj[d].c, adj[d].v, adj[d].nnz, 1.0f);

      // comb += 0.5 * B^T (B xt) for each applicable boundary
      const CooMat* bounds[2] = {bdown[d], bup[d]};
      for (int bi = 0; bi < 2; ++bi) {
        const CooMat* bm = bounds[bi];
        if (!bm) continue;
        fill_kernel<<<fillGrid, blk, 0, stream>>>(t1, 0.0f, NH);
        // t1 = B @ xt     : t1[r] += v * xt[c]
        spmm_scatter<<<(bm->nnz + 7) / 8, blk, 0, stream>>>(
            t1, xt, bm->r, bm->c, bm->v, bm->nnz, 1.0f);
        // comb += 0.5 * B^T @ t1 : comb[c] += 0.5 * v * t1[r]
        spmm_scatter<<<(bm->nnz + 7) / 8, blk, 0, stream>>>(
            comb, t1, bm->c, bm->r, bm->v, bm->nnz, 0.5f);
      }

      // enc_out = (d ? enc_out : 0) + (1/3) * sigmoid(comb@a) * comb
      gate_accum<<<gateGrid, blk, 0, stream>>>(
          comb, av, enc_out, N, 1.0f / 3.0f, d > 0 ? 1 : 0);
    }
  }

  // Predictor: h1 = PReLU(h_online @ pW1 + pb1); h_pred = h1 @ pW2 + pb2
  gemm_wmma_f32<HDIM, true ><<<gemmGrid, blk, 0, stream>>>(h_online, pW1, pb1, alpha, xt);
  gemm_wmma_f32<HDIM, false><<<gemmGrid, blk, 0, stream>>>(xt, pW2, pb2, nullptr, h_pred);
}